// EdgeGNNEncoder_49031346651547
// MI455X (gfx1250) — compile-verified
//
#include <hip/hip_runtime.h>
#include <cstddef>

#define N_NODES   50000
#define N_EDGES   800000
#define ET        850000            // edges + self loops
#define IN_CH     128
#define HID       64
#define HEADS     4
#define C1        (HEADS * HID)     // 256
#define NEG_SLOPE 0.2f
#define SM_EPS    1e-16f

typedef float v2f __attribute__((ext_vector_type(2)));
typedef float v8f __attribute__((ext_vector_type(8)));

// ---------- helpers ----------
__device__ __forceinline__ unsigned enc_f(float x) {
    unsigned b = __float_as_uint(x);
    return (b & 0x80000000u) ? ~b : (b | 0x80000000u);
}
__device__ __forceinline__ float dec_f(unsigned e) {
    unsigned b = (e & 0x80000000u) ? (e & 0x7FFFFFFFu) : ~e;
    return __uint_as_float(b);
}
__device__ __forceinline__ void edge_sd(const int* __restrict__ ei, int e, int& s, int& d) {
    if (e < N_EDGES) { s = ei[e]; d = ei[N_EDGES + e]; }
    else             { s = e - N_EDGES; d = s; }          // self loop
}

// ---------- init ----------
__global__ __launch_bounds__(256) void fill_f32(float* p, float v, int n) {
    int i = blockIdx.x * 256 + threadIdx.x;
    if (i < n) p[i] = v;
}
__global__ __launch_bounds__(256) void fill_u32(unsigned* p, unsigned v, int n) {
    int i = blockIdx.x * 256 + threadIdx.x;
    if (i < n) p[i] = v;
}

// ---------- WMMA f32 GEMM: D[M,N] = A[M,K] @ B[K,N] ----------
// One wave computes a 16 x (16*NT) strip, K-stepped by 4 (V_WMMA_F32_16X16X4_F32).
// A 16x4 frag loaded ONCE per K-step and reused across NT column tiles.
// A frag: lanes 0-15 hold K={0,1}, lanes 16-31 K={2,3}, M = lane&15.
// B frag: VGPR0/1 hold rows K={0,1} (half 0) / K={2,3} (half 1), N = lane&15.
// D: VGPR r -> row (r + 8*half), col = lane&15.
template <int NT>
__global__ __launch_bounds__(256) void gemm_wmma_f32(const float* __restrict__ A,
                                                     const float* __restrict__ B,
                                                     float* __restrict__ D,
                                                     int M, int K, int N) {
    const int lane = threadIdx.x & 31;
    const int wave = threadIdx.x >> 5;
    const int groupsN = (N >> 4) / NT;
    const int numGroups = (M >> 4) * groupsN;
    const int g = blockIdx.x * 8 + wave;
    if (g >= numGroups) return;                   // wave-uniform -> EXEC stays all-ones
    const int tm = g / groupsN;
    const int tn0 = (g - tm * groupsN) * NT;
    const int half = lane >> 4;
    const int l15 = lane & 15;
    const float* Arow = A + (size_t)(tm * 16 + l15) * K;
    const int col0 = tn0 * 16 + l15;
    v8f acc[NT] = {};
    for (int k = 0; k < K; k += 4) {
        const int ka = k + 2 * half;
        v2f a;
        a.x = Arow[ka];
        a.y = Arow[ka + 1];
        const float* Brow0 = B + (size_t)ka * N;
        const float* Brow1 = B + (size_t)(ka + 1) * N;
#pragma unroll
        for (int t = 0; t < NT; ++t) {
            v2f b;
            b.x = Brow0[col0 + 16 * t];
            b.y = Brow1[col0 + 16 * t];
            acc[t] = __builtin_amdgcn_wmma_f32_16x16x4_f32(false, a, false, b,
                                                           (short)0, acc[t],
                                                           false, false);
        }
    }
#pragma unroll
    for (int t = 0; t < NT; ++t)
#pragma unroll
        for (int r = 0; r < 8; ++r)
            D[(size_t)(tm * 16 + half * 8 + r) * N + col0 + 16 * t] = acc[t][r];
}

// ---------- attention dot products: a_s[n,h] = sum_c h[n,h,c]*att_s[h,c] ----------
template <int H>
__global__ __launch_bounds__(H * 64) void att_reduce(const float* __restrict__ hfeat,
                                                     const float* __restrict__ att_s,
                                                     const float* __restrict__ att_d,
                                                     float* __restrict__ a_s,
                                                     float* __restrict__ a_d) {
    const int n = blockIdx.x;
    const int t = threadIdx.x;                    // [0, H*64)
    __shared__ float ss[H * 64];
    __shared__ float sd[H * 64];
    const float hv = hfeat[(size_t)n * (H * 64) + t];
    ss[t] = hv * att_s[t];
    sd[t] = hv * att_d[t];
    __syncthreads();
    for (int off = 32; off >= 1; off >>= 1) {
        if ((t & 63) < off) { ss[t] += ss[t + off]; sd[t] += sd[t + off]; }
        __syncthreads();
    }
    if ((t & 63) == 0) {
        const int head = t >> 6;
        a_s[(size_t)n * H + head] = ss[t];
        a_d[(size_t)n * H + head] = sd[t];
    }
}

// ---------- edge pass A: logits + segment max ----------
template <int H>
__global__ __launch_bounds__(256) void edge_logits_max(const int* __restrict__ ei,
                                                       const float* __restrict__ a_s,
                                                       const float* __restrict__ a_d,
                                                       float* __restrict__ elog,
                                                       unsigned* __restrict__ emax) {
    const int idx = blockIdx.x * 256 + threadIdx.x;
    if (idx >= ET * H) return;
    const int e = idx / H, h = idx - e * H;
    int s, d; edge_sd(ei, e, s, d);
    float v = a_s[s * H + h] + a_d[d * H + h];
    v = (v >= 0.f) ? v : NEG_SLOPE * v;           // leaky_relu
    elog[idx] = v;
    atomicMax(&emax[d * H + h], enc_f(v));
}

// ---------- edge pass B: exp + segment sum ----------
template <int H>
__global__ __launch_bounds__(256) void edge_exp_sum(const int* __restrict__ ei,
                                                    float* __restrict__ elog,
                                                    const unsigned* __restrict__ emax,
                                                    float* __restrict__ denom) {
    const int idx = blockIdx.x * 256 + threadIdx.x;
    if (idx >= ET * H) return;
    const int e = idx / H, h = idx - e * H;
    int s, d; edge_sd(ei, e, s, d);
    const float m = dec_f(emax[d * H + h]);
    const float ex = expf(elog[idx] - m);
    elog[idx] = ex;                               // store numerator for pass C
    atomicAdd(&denom[d * H + h], ex);
}

// ---------- edge pass C: alpha-weighted scatter-add of messages ----------
template <int H>
__global__ __launch_bounds__(256) void edge_aggregate(const int* __restrict__ ei,
                                                      const float* __restrict__ hfeat,
                                                      const float* __restrict__ elog,
                                                      const float* __restrict__ denom,
                                                      float* __restrict__ outp) {
    constexpr int C = H * HID;                    // power of two
    const int idx = blockIdx.x * 256 + threadIdx.x;
    if (idx >= ET * C) return;
    const int e = idx / C;
    const int c = idx - e * C;
    const int h = c >> 6;
    int s, d; edge_sd(ei, e, s, d);
    const float alpha = elog[e * H + h] / (denom[d * H + h] + SM_EPS);
    atomicAdd(&outp[(size_t)d * C + c], hfeat[(size_t)s * C + c] * alpha);
}

// ---------- ELU(out + b1) in place (layer 1 -> layer 2 input) ----------
__global__ __launch_bounds__(256) void elu_bias1(float* __restrict__ p,
                                                 const float* __restrict__ b) {
    const int i = blockIdx.x * 256 + threadIdx.x;
    if (i >= N_NODES * C1) return;
    float v = p[i] + b[i & (C1 - 1)];
    p[i] = (v > 0.f) ? v : (expf(v) - 1.f);
}

// ---------- final gather: out[e] = concat(h2[src]+b2, h2[dst]+b2) ----------
__global__ __launch_bounds__(256) void gather_out(const int* __restrict__ ei,
                                                  const float* __restrict__ o2,
                                                  const float* __restrict__ b2,
                                                  float* __restrict__ out) {
    const int idx = blockIdx.x * 256 + threadIdx.x;
    if (idx >= N_EDGES * HID) return;
    const int e = idx >> 6, c = idx & 63;
    const int u = ei[e];
    const int v = ei[N_EDGES + e];
    const float bc = b2[c];
    out[(size_t)e * 128 + c]      = o2[(size_t)u * HID + c] + bc;
    out[(size_t)e * 128 + 64 + c] = o2[(size_t)v * HID + c] + bc;
}

extern "C" void kernel_launch(void* const* d_in, const int* in_sizes, int n_in,
                              void* d_out, int out_size, void* d_ws, size_t ws_size,
                              hipStream_t stream) {
    (void)in_sizes; (void)n_in; (void)out_size; (void)ws_size;
    const float* x    = (const float*)d_in[0];
    const int*   ei   = (const int*)  d_in[1];
    const float* W1   = (const float*)d_in[2];
    const float* atS1 = (const float*)d_in[3];
    const float* atD1 = (const float*)d_in[4];
    const float* b1   = (const float*)d_in[5];
    const float* W2   = (const float*)d_in[6];
    const float* atS2 = (const float*)d_in[7];
    const float* atD2 = (const float*)d_in[8];
    const float* b2   = (const float*)d_in[9];
    float* out = (float*)d_out;

    // workspace carve-up (256B aligned regions)
    char* w = (char*)d_ws;
    auto carve = [&](size_t nElem) {
        void* p = (void*)w;
        w += ((nElem * sizeof(float) + 255) & ~(size_t)255);
        return p;
    };
    float*    h1   = (float*)   carve((size_t)N_NODES * C1);
    float*    o1   = (float*)   carve((size_t)N_NODES * C1);
    float*    h2   = (float*)   carve((size_t)N_NODES * HID);
    float*    o2   = (float*)   carve((size_t)N_NODES * HID);
    float*    nas1 = (float*)   carve((size_t)N_NODES * HEADS);
    float*    nad1 = (float*)   carve((size_t)N_NODES * HEADS);
    unsigned* em1  = (unsigned*)carve((size_t)N_NODES * HEADS);
    float*    den1 = (float*)   carve((size_t)N_NODES * HEADS);
    float*    nas2 = (float*)   carve((size_t)N_NODES);
    float*    nad2 = (float*)   carve((size_t)N_NODES);
    unsigned* em2  = (unsigned*)carve((size_t)N_NODES);
    float*    den2 = (float*)   carve((size_t)N_NODES);
    float*    e1   = (float*)   carve((size_t)ET * HEADS);
    float*    e2   = (float*)   carve((size_t)ET);

    auto blocks = [](long long n) { return (unsigned)((n + 255) / 256); };

    // ---- init accumulators ----
    fill_f32<<<blocks((long long)N_NODES * C1), 256, 0, stream>>>(o1, 0.f, N_NODES * C1);
    fill_f32<<<blocks((long long)N_NODES * HID), 256, 0, stream>>>(o2, 0.f, N_NODES * HID);
    fill_f32<<<blocks(N_NODES * HEADS), 256, 0, stream>>>(den1, 0.f, N_NODES * HEADS);
    fill_f32<<<blocks(N_NODES), 256, 0, stream>>>(den2, 0.f, N_NODES);
    fill_u32<<<blocks(N_NODES * HEADS), 256, 0, stream>>>(em1, 0u, N_NODES * HEADS);
    fill_u32<<<blocks(N_NODES), 256, 0, stream>>>(em2, 0u, N_NODES);

    // ---- layer 1 ----
    {   // h1 = x @ W1   (M=50000, K=128, N=256): 3125 x 4 strip-groups of 16x64
        int numGroups = (N_NODES / 16) * ((C1 / 16) / 4);
        gemm_wmma_f32<4><<<(numGroups + 7) / 8, 256, 0, stream>>>(x, W1, h1,
                                                                  N_NODES, IN_CH, C1);
    }
    att_reduce<HEADS><<<N_NODES, HEADS * 64, 0, stream>>>(h1, atS1, atD1, nas1, nad1);
    edge_logits_max<HEADS><<<blocks((long long)ET * HEADS), 256, 0, stream>>>(
        ei, nas1, nad1, e1, em1);
    edge_exp_sum<HEADS><<<blocks((long long)ET * HEADS), 256, 0, stream>>>(
        ei, e1, em1, den1);
    edge_aggregate<HEADS><<<blocks((long long)ET * C1), 256, 0, stream>>>(
        ei, h1, e1, den1, o1);
    elu_bias1<<<blocks((long long)N_NODES * C1), 256, 0, stream>>>(o1, b1);

    // ---- layer 2 ----
    {   // h2 = elu(o1) @ W2  (M=50000, K=256, N=64): 3125 strip-groups of 16x64
        int numGroups = (N_NODES / 16) * ((HID / 16) / 4);
        gemm_wmma_f32<4><<<(numGroups + 7) / 8, 256, 0, stream>>>(o1, W2, h2,
                                                                  N_NODES, C1, HID);
    }
    att_reduce<1><<<N_NODES, 64, 0, stream>>>(h2, atS2, atD2, nas2, nad2);
    edge_logits_max<1><<<blocks(ET), 256, 0, stream>>>(ei, nas2, nad2, e2, em2);
    edge_exp_sum<1><<<blocks(ET), 256, 0, stream>>>(ei, e2, em2, den2);
    edge_aggregate<1><<<blocks((long long)ET * HID), 256, 0, stream>>>(
        ei, h2, e2, den2, o2);

    // ---- final edge gather (+ b2) ----
    gather_out<<<blocks((long long)N_EDGES * HID), 256, 0, stream>>>(ei, o2, b2, out);
}